// NN_386547056691
// MI455X (gfx1250) — compile-verified
//
#include <hip/hip_runtime.h>

typedef float v2f __attribute__((ext_vector_type(2)));
typedef float v8f __attribute__((ext_vector_type(8)));

namespace {
constexpr int F_   = 200;   // features
constexpr int DIN  = 3;     // input dim per feature
constexpr int H_   = 15;    // hidden
constexpr int ENC  = 30;    // encoder out per feature
constexpr int LDSW = 18;    // LDS row stride (dwords): even (b64 align), conflict-free
constexpr int ROWS_PER_WAVE = 16;   // rows per M-tile

// main kernel: 2 M-tiles per wave (32 rows), 4 waves/block -> 128 rows/block
constexpr int MTILES     = 2;
constexpr int WAVES_MAIN = 4;
constexpr int ROWS_MAIN  = ROWS_PER_WAVE * MTILES * WAVES_MAIN;   // 128
// fallback kernel: 8 waves/block, 1 tile
constexpr int WAVES_FB   = 8;
constexpr int ROWS_FB    = ROWS_PER_WAVE * WAVES_FB;              // 128

// Workspace layout (in floats)
constexpr int OFF_W1OP = 0;                         // [F][32 lanes][2]      = 12800
constexpr int OFF_W2OP = OFF_W1OP + F_ * 64;        // [F][4c][2t][32][2]    = 102400
constexpr int OFF_W3P  = OFF_W2OP + F_ * 512;       // [F][16 lanes][2]      = 6400
constexpr int WS_FLOATS = OFF_W3P + F_ * 32;        // 121600 floats = 486400 B

__device__ __forceinline__ v8f wmma_f32_k4(v2f a, v2f b, v8f c) {
  // D = A(16x4) * B(4x16) + C(16x16), fp32
  return __builtin_amdgcn_wmma_f32_16x16x4_f32(false, a, false, b,
                                               (short)0, c, false, false);
}
} // namespace

// ---------------------------------------------------------------------------
// Prep: pad weights into exact WMMA B-operand lane layout; fold b1 into the
// K=3 pad row of W1op and b2 into the K=15 pad row of W2op; pair up W3.
// ---------------------------------------------------------------------------
__global__ __launch_bounds__(256) void prep_weights(
    const float* __restrict__ W1, const float* __restrict__ b1,
    const float* __restrict__ W2, const float* __restrict__ b2,
    const float* __restrict__ W3, float* __restrict__ ws)
{
  const int f = blockIdx.x;
  const int t = threadIdx.x;

  // Layer-1 B operand (+ b1 folded): 64 floats per feature.
  if (t < 64) {
    const int lane = t >> 1, j = t & 1;
    const int half = lane >> 4, l16 = lane & 15;
    float v = 0.0f;
    if (l16 < H_) {
      if (j == 0)    v = W1[f * (DIN * H_) + (half ? 2 : 0) * H_ + l16];
      else if (half) v = b1[f * H_ + l16];                 // K=3 row := bias
      else           v = W1[f * (DIN * H_) + 1 * H_ + l16];
    }
    ws[OFF_W1OP + f * 64 + t] = v;
  }

  // Layer-2 B operands (+ b2 folded into K=15 row): 512 floats per feature.
  for (int k = t; k < 512; k += 256) {
    const int j    = k & 1;
    const int lane = (k >> 1) & 31;
    const int tt   = (k >> 6) & 1;
    const int c    = k >> 7;
    const int half = lane >> 4, l16 = lane & 15;
    const int kr   = 4 * c + 2 * half + j;
    const int col  = 16 * tt + l16;
    float v = 0.0f;
    if (col < ENC)
      v = (kr < H_) ? W2[f * (H_ * ENC) + kr * ENC + col]
                    : b2[f * ENC + col];                   // K=15 row := bias
    ws[OFF_W2OP + f * 512 + k] = v;
  }

  // W3 paired per lane: [f][l16][2] = { W3[f*30+l16], W3[f*30+16+l16] (pad 0) }
  if (t < 32) {
    const int l16 = t >> 1, j = t & 1;
    float v = 0.0f;
    if (j == 0)              v = W3[f * ENC + l16];
    else if (l16 < ENC - 16) v = W3[f * ENC + 16 + l16];
    ws[OFF_W3P + f * 32 + t] = v;
  }
}

// ---------------------------------------------------------------------------
// Main fused kernel: one wave = 2 x 16 batch rows, loops over all 200 features.
// ---------------------------------------------------------------------------
__global__ __launch_bounds__(WAVES_MAIN * 32) void fused_feature_mlp(
    const float* __restrict__ y,    // [B, F, 3]
    const float* __restrict__ ws,   // prepped operands
    const float* __restrict__ b3,   // [1]
    float* __restrict__ out,        // [B]
    int Btot)
{
  __shared__ float xpose[WAVES_MAIN][MTILES][ROWS_PER_WAVE * LDSW];

  const int lane = threadIdx.x & 31;   // wave32
  const int wave = threadIdx.x >> 5;
  const int half = lane >> 4;
  const int l16  = lane & 15;
  const int b0   = blockIdx.x * ROWS_MAIN + wave * (ROWS_PER_WAVE * MTILES);
  if (b0 >= Btot) return;

  float* tp0 = &xpose[wave][0][0];
  float* tp1 = &xpose[wave][1][0];
  const float* tpl0 = tp0 + l16 * LDSW + 2 * half;   // A-operand read bases
  const float* tpl1 = tp1 + l16 * LDSW + 2 * half;

  // One-time: pad column 15 of both tiles := 1.0 (feeds the b2 K=15 bias row).
  // lanes 0-15 cover tile0 rows, lanes 16-31 cover tile1 rows.
  {
    float* t = half ? tp1 : tp0;
    t[l16 * LDSW + 15] = 1.0f;
  }
  asm volatile("" ::: "memory");

  const float* yf0  = y + (size_t)(b0 + l16) * (F_ * DIN);
  const float* yf1  = yf0 + (size_t)ROWS_PER_WAVE * (F_ * DIN);
  const float* w1p  = ws + OFF_W1OP + lane * 2;
  const float* w2p  = ws + OFF_W2OP + lane * 2;
  const float* w3p2 = ws + OFF_W3P + l16 * 2;

  float acc0[8], acc1[8];
#pragma unroll
  for (int r = 0; r < 8; ++r) { acc0[r] = 0.0f; acc1[r] = 0.0f; }

  for (int f = 0; f < F_; ++f) {
    // ---- front-load ALL global loads (no deps -> max overlap) ----
    v2f bw2[8];
#pragma unroll
    for (int i = 0; i < 8; ++i) bw2[i] = *(const v2f*)(w2p + i * 64);
    const v2f w3v = *(const v2f*)w3p2;
    const v2f bw1 = *(const v2f*)w1p;

    v2f a10, a11;
    a10.x = yf0[half ? 2 : 0];
    a11.x = yf1[half ? 2 : 0];
    const float y01 = yf0[1];
    const float y11 = yf1[1];
    a10.y = half ? 1.0f : y01;
    a11.y = half ? 1.0f : y11;

    // ---- Layer 1 (bias folded into K=3), two independent M-tiles ----
    v8f c1t0 = {}, c1t1 = {};
    c1t0 = wmma_f32_k4(a10, bw1, c1t0);
    c1t1 = wmma_f32_k4(a11, bw1, c1t1);

    // ---- ReLU + transpose (C layout -> A layout) via per-wave LDS tiles.
    // Column 15 is statically 1.0 (pre-initialized) and never rewritten.
    if (l16 < 15) {
#pragma unroll
      for (int r = 0; r < 8; ++r) {
        tp0[(r + 8 * half) * LDSW + l16] = __builtin_fmaxf(c1t0[r], 0.0f);
        tp1[(r + 8 * half) * LDSW + l16] = __builtin_fmaxf(c1t1[r], 0.0f);
      }
    }
    asm volatile("" ::: "memory");     // DS in-order per wave; stop reordering

    // ---- Layer 2 (bias folded into K=15): 4 K-chunks x 2 N-tiles x 2 M-tiles ----
    v8f c2a0 = {}, c2b0 = {}, c2a1 = {}, c2b1 = {};
#pragma unroll
    for (int c = 0; c < 4; ++c) {
      const v2f a20 = *(const v2f*)(tpl0 + 4 * c);
      const v2f a21 = *(const v2f*)(tpl1 + 4 * c);
      c2a0 = wmma_f32_k4(a20, bw2[2 * c + 0], c2a0);
      c2a1 = wmma_f32_k4(a21, bw2[2 * c + 0], c2a1);
      c2b0 = wmma_f32_k4(a20, bw2[2 * c + 1], c2b0);
      c2b1 = wmma_f32_k4(a21, bw2[2 * c + 1], c2b1);
    }

    // ---- ReLU + fold into final dot product ----
#pragma unroll
    for (int r = 0; r < 8; ++r) {
      acc0[r] = fmaf(__builtin_fmaxf(c2a0[r], 0.0f), w3v.x, acc0[r]);
      acc0[r] = fmaf(__builtin_fmaxf(c2b0[r], 0.0f), w3v.y, acc0[r]);
      acc1[r] = fmaf(__builtin_fmaxf(c2a1[r], 0.0f), w3v.x, acc1[r]);
      acc1[r] = fmaf(__builtin_fmaxf(c2b1[r], 0.0f), w3v.y, acc1[r]);
    }

    yf0 += DIN; yf1 += DIN; w1p += 64; w2p += 512; w3p2 += 32;
  }

  // ---- reduce over the 16 column-lanes in each half ----
#pragma unroll
  for (int r = 0; r < 8; ++r) {
    float v0 = acc0[r];
    float v1 = acc1[r];
    v0 += __shfl_xor(v0, 8, 32);  v1 += __shfl_xor(v1, 8, 32);
    v0 += __shfl_xor(v0, 4, 32);  v1 += __shfl_xor(v1, 4, 32);
    v0 += __shfl_xor(v0, 2, 32);  v1 += __shfl_xor(v1, 2, 32);
    v0 += __shfl_xor(v0, 1, 32);  v1 += __shfl_xor(v1, 1, 32);
    acc0[r] = v0; acc1[r] = v1;
  }

  if (l16 == 0) {
    const float bb3 = b3[0];
#pragma unroll
    for (int r = 0; r < 8; ++r) {
      const int row0 = b0 + r + 8 * half;
      const int row1 = row0 + ROWS_PER_WAVE;
      if (row0 < Btot) out[row0] = acc0[r] + bb3;
      if (row1 < Btot) out[row1] = acc1[r] + bb3;
    }
  }
}

// ---------------------------------------------------------------------------
// Fallback (no workspace): kernel with predicated loads.
// ---------------------------------------------------------------------------
__global__ __launch_bounds__(WAVES_FB * 32) void fused_feature_mlp_fallback(
    const float* __restrict__ y, const float* __restrict__ W1,
    const float* __restrict__ b1, const float* __restrict__ W2,
    const float* __restrict__ b2, const float* __restrict__ W3,
    const float* __restrict__ b3, float* __restrict__ out, int Btot)
{
  __shared__ float xpose[WAVES_FB][ROWS_PER_WAVE * LDSW];
  const int lane = threadIdx.x & 31;
  const int wave = threadIdx.x >> 5;
  const int half = lane >> 4;
  const int l16  = lane & 15;
  const int b0   = blockIdx.x * ROWS_FB + wave * ROWS_PER_WAVE;
  if (b0 >= Btot) return;

  float* tp = &xpose[wave][0];
  const float* yrow = y + (size_t)(b0 + l16) * (F_ * DIN);
  float acc[8];
#pragma unroll
  for (int r = 0; r < 8; ++r) acc[r] = 0.0f;

  for (int f = 0; f < F_; ++f) {
    const float* yf = yrow + f * DIN;
    v2f a1;
    a1.x = yf[half ? 2 : 0];
    const float y1 = yf[1];
    a1.y = half ? 0.0f : y1;
    const float* w1f = W1 + f * (DIN * H_);
    const bool ncol1 = (l16 < H_);
    v2f bw1;
    bw1.x = ncol1 ? w1f[(half ? 2 : 0) * H_ + l16] : 0.0f;
    bw1.y = (ncol1 && !half) ? w1f[1 * H_ + l16] : 0.0f;
    v8f c1 = {};
    c1 = wmma_f32_k4(a1, bw1, c1);
    const float bias1 = ncol1 ? b1[f * H_ + l16] : 0.0f;
#pragma unroll
    for (int r = 0; r < 8; ++r)
      c1[r] = __builtin_fmaxf(c1[r] + bias1, 0.0f);
#pragma unroll
    for (int r = 0; r < 8; ++r)
      tp[(r + 8 * half) * LDSW + l16] = c1[r];
    asm volatile("" ::: "memory");
    const float* w2f = W2 + f * (H_ * ENC);
    v8f c2a = {}, c2b = {};
#pragma unroll
    for (int c = 0; c < 4; ++c) {
      const v2f a2 = *(const v2f*)(tp + l16 * LDSW + 4 * c + 2 * half);
      const int krow = 4 * c + 2 * half;
#pragma unroll
      for (int t = 0; t < 2; ++t) {
        const int col = 16 * t + l16;
        const bool cv = (col < ENC);
        v2f bw2;
        bw2.x = (cv && krow < H_)       ? w2f[krow * ENC + col]       : 0.0f;
        bw2.y = (cv && (krow + 1) < H_) ? w2f[(krow + 1) * ENC + col] : 0.0f;
        if (t == 0) c2a = wmma_f32_k4(a2, bw2, c2a);
        else        c2b = wmma_f32_k4(a2, bw2, c2b);
      }
    }
#pragma unroll
    for (int t = 0; t < 2; ++t) {
      const int col = 16 * t + l16;
      const bool cv = (col < ENC);
      const float bias2 = cv ? b2[f * ENC + col] : 0.0f;
      const float w3v   = cv ? W3[f * ENC + col] : 0.0f;
#pragma unroll
      for (int r = 0; r < 8; ++r) {
        const float v = __builtin_fmaxf((t == 0 ? c2a[r] : c2b[r]) + bias2, 0.0f);
        acc[r] = fmaf(v, w3v, acc[r]);
      }
    }
  }
#pragma unroll
  for (int r = 0; r < 8; ++r) {
    float v = acc[r];
    v += __shfl_xor(v, 8, 32);
    v += __shfl_xor(v, 4, 32);
    v += __shfl_xor(v, 2, 32);
    v += __shfl_xor(v, 1, 32);
    acc[r] = v;
  }
  if (l16 == 0) {
    const float bb3 = b3[0];
#pragma unroll
    for (int r = 0; r < 8; ++r) {
      const int row = b0 + r + 8 * half;
      if (row < Btot) out[row] = acc[r] + bb3;
    }
  }
}

extern "C" void kernel_launch(void* const* d_in, const int* in_sizes, int n_in,
                              void* d_out, int out_size, void* d_ws, size_t ws_size,
                              hipStream_t stream) {
  const float* y  = (const float*)d_in[0];
  const float* W1 = (const float*)d_in[1];
  const float* b1 = (const float*)d_in[2];
  const float* W2 = (const float*)d_in[3];
  const float* b2 = (const float*)d_in[4];
  const float* W3 = (const float*)d_in[5];
  const float* b3 = (const float*)d_in[6];
  float* out = (float*)d_out;

  const int Btot = in_sizes[0] / (F_ * DIN);                   // 32768

  if (ws_size >= (size_t)WS_FLOATS * sizeof(float)) {
    float* ws = (float*)d_ws;
    const int grid = (Btot + ROWS_MAIN - 1) / ROWS_MAIN;       // 256
    prep_weights<<<F_, 256, 0, stream>>>(W1, b1, W2, b2, W3, ws);
    fused_feature_mlp<<<grid, WAVES_MAIN * 32, 0, stream>>>(y, ws, b3, out, Btot);
  } else {
    const int grid = (Btot + ROWS_FB - 1) / ROWS_FB;           // 256
    fused_feature_mlp_fallback<<<grid, WAVES_FB * 32, 0, stream>>>(
        y, W1, b1, W2, b2, W3, b3, out, Btot);
  }
}